// Decoder_62079457296866
// MI455X (gfx1250) — compile-verified
//
#include <hip/hip_runtime.h>
#include <hip/hip_bf16.h>

#define B_SZ   128
#define N_ATT  49
#define D_DIM  512
#define H_DIM  512
#define V_SZ   10000
#define E_DIM  512
#define T_LEN  20
#define KCAT   1536   // D + E + H  (context | emb | h)
#define G4H    2048   // 4*H gates

typedef __attribute__((ext_vector_type(2))) float v2f;
typedef __attribute__((ext_vector_type(8))) float v8f;

#define WMMA_F32(A, B, C) \
    __builtin_amdgcn_wmma_f32_16x16x4_f32(false, (A), false, (B), (short)0, (C), false, false)

// ---------------------------------------------------------------------------
// Generic C[M,N] = X[M,K] @ W[N,K]^T (+bias0 +bias1) via V_WMMA_F32_16X16X4_F32
// Wave computes a 32x32 tile (2x2 WMMA tiles); block = 4 waves -> 64x64 tile.
// M must be a multiple of 64; N a multiple of 16 (col tiles guarded uniformly).
// Software-pipelined: chunk k+4 fragments are loaded before chunk k's WMMAs so
// the loadcnt wait lands after the matrix ops of the previous chunk.
// ---------------------------------------------------------------------------
template<bool TWO_COLS>
__device__ __forceinline__ void gemm_core(
    const float* __restrict__ X, const float* __restrict__ W, int K,
    int row0, int col0, int r, int half,
    v8f& acc00, v8f& acc01, v8f& acc10, v8f& acc11)
{
    const float* xa0 = X + (size_t)(row0 + r)      * K + 2 * half;
    const float* xa1 = X + (size_t)(row0 + 16 + r) * K + 2 * half;
    const float* wb0 = W + (size_t)(col0 + r)      * K + 2 * half;
    const float* wb1 = W + (size_t)(col0 + 16 + r) * K + 2 * half;

    v2f a0 = *(const v2f*)(xa0);
    v2f a1 = *(const v2f*)(xa1);
    v2f b0 = *(const v2f*)(wb0);
    v2f b1;
    if (TWO_COLS) b1 = *(const v2f*)(wb1);

#pragma unroll 2
    for (int k = 4; k < K; k += 4) {
        // prefetch next chunk's fragments (independent of current WMMAs)
        v2f na0 = *(const v2f*)(xa0 + k);
        v2f na1 = *(const v2f*)(xa1 + k);
        v2f nb0 = *(const v2f*)(wb0 + k);
        v2f nb1;
        if (TWO_COLS) nb1 = *(const v2f*)(wb1 + k);

        acc00 = WMMA_F32(a0, b0, acc00);
        acc10 = WMMA_F32(a1, b0, acc10);
        if (TWO_COLS) {
            acc01 = WMMA_F32(a0, b1, acc01);
            acc11 = WMMA_F32(a1, b1, acc11);
        }
        a0 = na0; a1 = na1; b0 = nb0;
        if (TWO_COLS) b1 = nb1;
    }
    // epilogue chunk
    acc00 = WMMA_F32(a0, b0, acc00);
    acc10 = WMMA_F32(a1, b0, acc10);
    if (TWO_COLS) {
        acc01 = WMMA_F32(a0, b1, acc01);
        acc11 = WMMA_F32(a1, b1, acc11);
    }
}

__global__ void gemm_nt_kernel(const float* __restrict__ X,
                               const float* __restrict__ W,
                               float* __restrict__ C,
                               const float* __restrict__ bias0,
                               const float* __restrict__ bias1,
                               int M, int N, int K, int remapT, int remapB)
{
    const int lane = threadIdx.x & 31;
    const int wave = threadIdx.x >> 5;              // 0..3
    const int half = lane >> 4;                     // 0|1
    const int r    = lane & 15;
    const int row0 = blockIdx.y * 64 + (wave >> 1) * 32;
    const int col0 = blockIdx.x * 64 + (wave & 1) * 32;
    if (col0 + 16 > N) return;                      // whole wave out of range (uniform)
    const bool two = (col0 + 32) <= N;

    v8f acc00 = {}, acc01 = {}, acc10 = {}, acc11 = {};
    if (two) gemm_core<true >(X, W, K, row0, col0, r, half, acc00, acc01, acc10, acc11);
    else     gemm_core<false>(X, W, K, row0, col0, r, half, acc00, acc01, acc10, acc11);

    const int cc0 = col0 + r;
    const int cc1 = cc0 + 16;
    float bA = (bias0 ? bias0[cc0] : 0.f) + (bias1 ? bias1[cc0] : 0.f);
    float bB = two ? ((bias0 ? bias0[cc1] : 0.f) + (bias1 ? bias1[cc1] : 0.f)) : 0.f;

#pragma unroll
    for (int v = 0; v < 8; ++v) {
        int rr0 = row0 + v + 8 * half;
        int rr1 = rr0 + 16;
        size_t o00, o01, o10, o11;
        if (remapB > 0) {   // row m = t*remapB + b  ->  [(b*remapT + t)*N + col]
            int t0 = rr0 / remapB, b0 = rr0 % remapB;
            int t1 = rr1 / remapB, b1 = rr1 % remapB;
            o00 = ((size_t)b0 * remapT + t0) * N + cc0;
            o01 = ((size_t)b0 * remapT + t0) * N + cc1;
            o10 = ((size_t)b1 * remapT + t1) * N + cc0;
            o11 = ((size_t)b1 * remapT + t1) * N + cc1;
        } else {
            o00 = (size_t)rr0 * N + cc0;  o01 = (size_t)rr0 * N + cc1;
            o10 = (size_t)rr1 * N + cc0;  o11 = (size_t)rr1 * N + cc1;
        }
        C[o00] = acc00[v] + bA;
        C[o10] = acc10[v] + bA;
        if (two) { C[o01] = acc01[v] + bB; C[o11] = acc11[v] + bB; }
    }
}

// ---------------------------------------------------------------------------
// mean_ctx[b,d] = mean_n img[b,n,d]
// ---------------------------------------------------------------------------
__global__ void mean_kernel(const float* __restrict__ img, float* __restrict__ meanc)
{
    int idx = blockIdx.x * blockDim.x + threadIdx.x;
    if (idx >= B_SZ * D_DIM) return;
    int b = idx / D_DIM, d = idx % D_DIM;
    const float* p = img + ((size_t)b * N_ATT) * D_DIM + d;
    float s = 0.f;
    for (int n = 0; n < N_ATT; ++n) s += p[(size_t)n * D_DIM];
    meanc[idx] = s * (1.0f / N_ATT);
}

// ---------------------------------------------------------------------------
// W_cat[j, 0:1024] = W_ih[j,:];  W_cat[j, 1024:1536] = W_hh[j,:]
// ---------------------------------------------------------------------------
__global__ void wcat_kernel(const float* __restrict__ W_ih,
                            const float* __restrict__ W_hh,
                            float* __restrict__ Wcat)
{
    int idx = blockIdx.x * blockDim.x + threadIdx.x;
    if (idx >= G4H * KCAT) return;
    int j = idx / KCAT, k = idx % KCAT;
    Wcat[idx] = (k < (E_DIM + D_DIM)) ? W_ih[(size_t)j * (E_DIM + D_DIM) + k]
                                      : W_hh[(size_t)j * H_DIM + (k - (E_DIM + D_DIM))];
}

__global__ void zero_kernel(float* __restrict__ p, int n)
{
    int idx = blockIdx.x * blockDim.x + threadIdx.x;
    if (idx < n) p[idx] = 0.f;
}

// ---------------------------------------------------------------------------
// Attention + build x = [context | emb_t | h] row, write alphas[:, :, t].
// One block (256 thr = 8 waves) per batch element.
// ---------------------------------------------------------------------------
__global__ void att_kernel(const float* __restrict__ fv,     // [B,N,D]
                           const float* __restrict__ img,    // [B,N,D]
                           const float* __restrict__ oh,     // [B,D]
                           const float* __restrict__ att_b,  // [N]
                           const float* __restrict__ W_a,    // [D]
                           const float* __restrict__ meanc,  // [B,D]
                           const float* __restrict__ h,      // [B,H]
                           const float* __restrict__ embed,  // [V,E]
                           const int*   __restrict__ captions, // [B,T]
                           float* __restrict__ xcat,         // [B,KCAT]
                           float* __restrict__ alphas,       // [B,N,T]
                           int t)
{
    const int b    = blockIdx.x;
    const int tid  = threadIdx.x;
    const int lane = tid & 31;
    const int wave = tid >> 5;
    __shared__ float sE[64];
    __shared__ float sA[64];

    // attention scores e[n] = sum_d relu(fv + oh + att_b[n]) * W_a[d]
    const float* ohp = oh + (size_t)b * D_DIM;
    for (int n = wave; n < N_ATT; n += 8) {
        const float* fvp = fv + ((size_t)b * N_ATT + n) * D_DIM;
        float ab = att_b[n];
        float s = 0.f;
        for (int d = lane; d < D_DIM; d += 32) {
            float v = fvp[d] + ohp[d] + ab;
            v = v > 0.f ? v : 0.f;
            s += v * W_a[d];
        }
        for (int off = 16; off > 0; off >>= 1) s += __shfl_xor(s, off, 32);
        if (lane == 0) sE[n] = s;
    }
    __syncthreads();

    // stable softmax over 49 (redundant per-thread; tiny)
    float mx = -3.4e38f;
    for (int n = 0; n < N_ATT; ++n) mx = fmaxf(mx, sE[n]);
    float ssum = 0.f;
    for (int n = 0; n < N_ATT; ++n) ssum += expf(sE[n] - mx);
    float inv = 1.f / ssum;
    if (tid < N_ATT) {
        float a = expf(sE[tid] - mx) * inv;
        sA[tid] = a;
        alphas[((size_t)b * N_ATT + tid) * T_LEN + t] = (t == 0) ? 0.f : a;
    }
    __syncthreads();

    // context (mean_ctx at t==0) and x = [ctx | emb | h]
    const int cap = captions[b * T_LEN + t];
    const float* embp = embed + (size_t)cap * E_DIM;
    float* xr = xcat + (size_t)b * KCAT;
    for (int d = tid; d < D_DIM; d += 256) {
        const float* ip = img + ((size_t)b * N_ATT) * D_DIM + d;
        float z = 0.f;
        for (int n = 0; n < N_ATT; ++n) z += ip[(size_t)n * D_DIM] * sA[n];
        float ctx = (t == 0) ? meanc[(size_t)b * D_DIM + d] : z;
        xr[d]                 = ctx;
        xr[D_DIM + d]         = embp[d];
        xr[D_DIM + E_DIM + d] = h[(size_t)b * H_DIM + d];
    }
}

// ---------------------------------------------------------------------------
// LSTM cell pointwise update; stores h_t into H_all[t*B + b].
// ---------------------------------------------------------------------------
__global__ void lstm_cell_kernel(const float* __restrict__ gates, // [B,4H]
                                 float* __restrict__ c,           // [B,H]
                                 float* __restrict__ h,           // [B,H]
                                 float* __restrict__ Hall,        // [T*B,H]
                                 int t)
{
    int idx = blockIdx.x * blockDim.x + threadIdx.x;
    if (idx >= B_SZ * H_DIM) return;
    int b = idx >> 9;           // / 512
    int j = idx & 511;
    const float* g = gates + (size_t)b * G4H;
    float gi = g[j], gf = g[H_DIM + j], gg = g[2 * H_DIM + j], go = g[3 * H_DIM + j];
    float si = 1.f / (1.f + expf(-gi));
    float sf = 1.f / (1.f + expf(-gf));
    float so = 1.f / (1.f + expf(-go));
    float cn = sf * c[idx] + si * tanhf(gg);
    float hn = so * tanhf(cn);
    c[idx] = cn;
    h[idx] = hn;
    Hall[((size_t)t * B_SZ + b) * H_DIM + j] = hn;
}

// ---------------------------------------------------------------------------
extern "C" void kernel_launch(void* const* d_in, const int* in_sizes, int n_in,
                              void* d_out, int out_size, void* d_ws, size_t ws_size,
                              hipStream_t stream)
{
    const float* img   = (const float*)d_in[0];   // [B,N,D]
    const int*   caps  = (const int*)  d_in[1];   // [B,T]
    const float* embed = (const float*)d_in[2];   // [V,E]
    const float* W_v   = (const float*)d_in[3];   // [D,D]
    const float* W_h   = (const float*)d_in[4];   // [D,H]
    const float* att_b = (const float*)d_in[5];   // [N]
    const float* W_a   = (const float*)d_in[6];   // [D]
    const float* W_ih  = (const float*)d_in[7];   // [4H, E+D]
    const float* b_ih  = (const float*)d_in[8];   // [4H]
    const float* W_hh  = (const float*)d_in[9];   // [4H, H]
    const float* b_hh  = (const float*)d_in[10];  // [4H]
    const float* W_out = (const float*)d_in[11];  // [V,H]
    const float* b_out = (const float*)d_in[12];  // [V]

    float* pred   = (float*)d_out;                               // [B,T,V]
    float* alphas = pred + (size_t)B_SZ * T_LEN * V_SZ;          // [B,N,T]

    // workspace layout
    float* ws    = (float*)d_ws;
    float* fv    = ws;                                           // B*N*D
    float* meanc = fv    + (size_t)B_SZ * N_ATT * D_DIM;         // B*D
    float* ohb   = meanc + (size_t)B_SZ * D_DIM;                 // B*D
    float* hcur  = ohb   + (size_t)B_SZ * D_DIM;                 // B*H
    float* ccur  = hcur  + (size_t)B_SZ * H_DIM;                 // B*H
    float* xcat  = ccur  + (size_t)B_SZ * H_DIM;                 // B*KCAT
    float* gates = xcat  + (size_t)B_SZ * KCAT;                  // B*4H
    float* Hall  = gates + (size_t)B_SZ * G4H;                   // T*B*H
    float* Wcat  = Hall  + (size_t)T_LEN * B_SZ * H_DIM;         // 4H*KCAT

    dim3 blk(128);

    // prep: concatenated gate weights, mean context, zero h/c
    {
        int n = G4H * KCAT;
        wcat_kernel<<<(n + 255) / 256, 256, 0, stream>>>(W_ih, W_hh, Wcat);
        mean_kernel<<<(B_SZ * D_DIM + 255) / 256, 256, 0, stream>>>(img, meanc);
        zero_kernel<<<(2 * B_SZ * H_DIM + 255) / 256, 256, 0, stream>>>(hcur, 2 * B_SZ * H_DIM);
    }

    // fv[b,n,e] = sum_d img[b,n,d] * W_v[e,d]   : [6272,512] x [512,512]^T
    {
        dim3 grid((D_DIM + 63) / 64, (B_SZ * N_ATT + 63) / 64);
        gemm_nt_kernel<<<grid, blk, 0, stream>>>(img, W_v, fv, nullptr, nullptr,
                                                 B_SZ * N_ATT, D_DIM, D_DIM, 0, 0);
    }

    // recurrent steps
    for (int t = 0; t < T_LEN; ++t) {
        // oh = h @ W_h^T : [128,512] x [512,512]^T
        {
            dim3 grid((D_DIM + 63) / 64, (B_SZ + 63) / 64);
            gemm_nt_kernel<<<grid, blk, 0, stream>>>(hcur, W_h, ohb, nullptr, nullptr,
                                                     B_SZ, D_DIM, H_DIM, 0, 0);
        }
        // attention + x row assembly + alphas
        att_kernel<<<B_SZ, 256, 0, stream>>>(fv, img, ohb, att_b, W_a, meanc,
                                             hcur, embed, caps, xcat, alphas, t);
        // gates = x @ Wcat^T + b_ih + b_hh : [128,1536] x [2048,1536]^T
        {
            dim3 grid((G4H + 63) / 64, (B_SZ + 63) / 64);
            gemm_nt_kernel<<<grid, blk, 0, stream>>>(xcat, Wcat, gates, b_ih, b_hh,
                                                     B_SZ, G4H, KCAT, 0, 0);
        }
        // LSTM pointwise
        lstm_cell_kernel<<<(B_SZ * H_DIM + 255) / 256, 256, 0, stream>>>(
            gates, ccur, hcur, Hall, t);
    }

    // predictions = Hall @ W_out^T + b_out : [2560,512] x [10000,512]^T
    // rows are (t*B+b); remap writes to pred[b,t,:]
    {
        dim3 grid((V_SZ + 63) / 64, (T_LEN * B_SZ + 63) / 64);
        gemm_nt_kernel<<<grid, blk, 0, stream>>>(Hall, W_out, pred, b_out, nullptr,
                                                 T_LEN * B_SZ, V_SZ, H_DIM,
                                                 T_LEN, B_SZ);
    }
}